// FastFlowDecoder_28913719836683
// MI455X (gfx1250) — compile-verified
//
#include <hip/hip_runtime.h>

typedef __attribute__((ext_vector_type(2))) float v2f;
typedef __attribute__((ext_vector_type(8))) float v8f;

#define B_   4
#define C_   64
#define H_   512
#define W_   512
#define HW_  ((size_t)H_ * (size_t)W_)
#define N_PTS 100000
#define PTS_PER_WAVE 16
#define WAVES_PER_BLOCK 8
#define PTS_PER_BLOCK (PTS_PER_WAVE * WAVES_PER_BLOCK)

__global__ __launch_bounds__(256) void flow_decoder_kernel(
    const float* __restrict__ before, const float* __restrict__ after,
    const int*   __restrict__ vc,     const float* __restrict__ W1,
    const float* __restrict__ b1,     const float* __restrict__ W2,
    const float* __restrict__ b2,     float* __restrict__ out,
    int tilesPerBatch)
{
    // W1 pre-swizzled into WMMA B-fragment order:
    // sW1p[(kc*2 + hp)*32 + n] = ( W1[(4*kc+2*hp  )*32 + n],
    //                              W1[(4*kc+2*hp+1)*32 + n] )
    // so each lane's B fragment is ONE aligned ds_load_b64 into an even VGPR pair.
    __shared__ v2f   sW1p[32 * 2 * 32];             // 16 KB
    __shared__ float sW2[32 * 3];
    __shared__ float sB1[32];
    __shared__ float sB2[3];
    __shared__ float sH[WAVES_PER_BLOCK][16][32];   // 16 KB, per-wave h tiles

    const int tid = threadIdx.x;
    #pragma unroll
    for (int it = 0; it < 8; ++it) {                // 2048 pairs / 256 threads
        const int i  = tid + it * 256;
        const int kc = i >> 6;                      // 0..31
        const int hp = (i >> 5) & 1;                // 0..1  (khalf/2)
        const int n  = i & 31;                      // 0..31
        const int ch0 = 4 * kc + 2 * hp;
        v2f p;
        p.x = W1[ch0 * 32 + n];
        p.y = W1[ch0 * 32 + 32 + n];
        sW1p[i] = p;
    }
    if (tid < 96) sW2[tid] = W2[tid];
    if (tid < 32) sB1[tid] = b1[tid];
    if (tid < 3)  sB2[tid] = b2[tid];
    __syncthreads();

    const int b     = blockIdx.x / tilesPerBatch;
    const int tile  = blockIdx.x % tilesPerBatch;
    const int wave  = tid >> 5;
    const int lane  = tid & 31;
    const int row   = lane & 15;            // point row within 16-pt tile (also N col for B/C frags)
    const int hp    = lane >> 4;            // which K pair of each 4-wide chunk
    const int khalf = hp * 2;

    const int p  = tile * PTS_PER_BLOCK + wave * PTS_PER_WAVE + row;
    const int pc = (p < N_PTS) ? p : (N_PTS - 1);   // clamp tail for gather

    const int y = vc[((size_t)b * N_PTS + pc) * 3 + 1];
    const int x = vc[((size_t)b * N_PTS + pc) * 3 + 2];

    const size_t pix   = (size_t)y * W_ + (size_t)x;
    const float* base0 = before + (size_t)b * C_ * HW_ + pix;
    const float* base1 = after  + (size_t)b * C_ * HW_ + pix;

    // ---- Gather: 64 independent scattered dword loads per lane, issued up front.
    // A-fragment layout of V_WMMA_F32_16X16X4_F32: lanes 0-15 hold K=0/1,
    // lanes 16-31 hold K=2/3 of each 4-wide K chunk, M = lane&15.
    v2f afrag[32];
    #pragma unroll
    for (int kc = 0; kc < 32; ++kc) {
        const int ch0 = 4 * kc + khalf;             // even; never straddles the 64-channel boundary
        const float* src = (ch0 < 64) ? (base0 + (size_t)ch0 * HW_)
                                      : (base1 + (size_t)(ch0 - 64) * HW_);
        v2f a;
        a.x = src[0];
        a.y = src[HW_];
        afrag[kc] = a;
    }

    // ---- Layer 1: h(16x32) = feat(16x128) @ W1(128x32) + b1 via f32 WMMA.
    #pragma unroll
    for (int half = 0; half < 2; ++half) {
        const int n = half * 16 + row;              // output column this lane holds
        const float bias = sB1[n];
        v8f acc;
        #pragma unroll
        for (int v = 0; v < 8; ++v) acc[v] = bias;  // D = A@B + C, C preloaded with b1

        #pragma unroll
        for (int kc = 0; kc < 32; ++kc) {
            const v2f bfrag = sW1p[(kc * 2 + hp) * 32 + n];  // one ds_load_b64
            acc = __builtin_amdgcn_wmma_f32_16x16x4_f32(
                /*neg_a=*/false, afrag[kc],
                /*neg_b=*/false, bfrag,
                /*c_mod=*/(short)0, acc,
                /*reuse_a=*/false, /*reuse_b=*/false);
        }

        // C/D layout: VGPR v, lane -> M = v + 8*(lane>>4), N = lane&15 (+half*16)
        #pragma unroll
        for (int v = 0; v < 8; ++v) {
            const int mrow = v + 8 * hp;
            sH[wave][mrow][n] = acc[v];
        }
    }
    __syncthreads();

    // ---- Layer 2: out(16x3) = h(16x32) @ W2(32x3) + b2; 96 FMAs per point.
    if (lane < 16 && p < N_PTS) {
        float o0 = sB2[0], o1 = sB2[1], o2 = sB2[2];
        #pragma unroll
        for (int n2 = 0; n2 < 32; ++n2) {
            const float hv = sH[wave][row][n2];
            o0 += hv * sW2[n2 * 3 + 0];
            o1 += hv * sW2[n2 * 3 + 1];
            o2 += hv * sW2[n2 * 3 + 2];
        }
        float* o = out + ((size_t)b * N_PTS + p) * 3;
        o[0] = o0; o[1] = o1; o[2] = o2;
    }
}

extern "C" void kernel_launch(void* const* d_in, const int* in_sizes, int n_in,
                              void* d_out, int out_size, void* d_ws, size_t ws_size,
                              hipStream_t stream) {
    const float* before = (const float*)d_in[0];
    const float* after  = (const float*)d_in[1];
    const int*   vc     = (const int*)  d_in[2];
    const float* W1     = (const float*)d_in[3];
    const float* b1     = (const float*)d_in[4];
    const float* W2     = (const float*)d_in[5];
    const float* b2     = (const float*)d_in[6];
    float* out = (float*)d_out;

    const int tilesPerBatch = (N_PTS + PTS_PER_BLOCK - 1) / PTS_PER_BLOCK;  // 782
    dim3 grid(B_ * tilesPerBatch);
    flow_decoder_kernel<<<grid, 256, 0, stream>>>(before, after, vc, W1, b1, W2, b2,
                                                  out, tilesPerBatch);
}